// TopoteinModelV0_26809185862173
// MI455X (gfx1250) — compile-verified
//
#include <hip/hip_runtime.h>

typedef __attribute__((ext_vector_type(2))) float v2f;
typedef __attribute__((ext_vector_type(8))) float v8f;

#if defined(__has_builtin)
#  if __has_builtin(__builtin_amdgcn_wmma_f32_16x16x4_f32)
#    define HAVE_WMMA_F32 1
#  endif
#endif
#ifndef HAVE_WMMA_F32
#  define HAVE_WMMA_F32 0
#endif

__device__ __forceinline__ v8f wmma_f32_16x16x4(v2f a, v2f b, v8f c) {
#if HAVE_WMMA_F32
  // 8 args: (neg_a, A, neg_b, B, c_mod, C, reuse_a, reuse_b)
  return __builtin_amdgcn_wmma_f32_16x16x4_f32(false, a, false, b, (short)0, c, false, false);
#else
  c[0] += a.x * b.x + a.y * b.y;  // placeholder if builtin absent (histogram reveals)
  return c;
#endif
}

// gfx1250 async global->LDS copy (16B per lane), tracked by ASYNCcnt.
__device__ __forceinline__ void async_g2l_b128(unsigned lds_off, const void* gptr) {
  asm volatile("global_load_async_to_lds_b128 %0, %1, off"
               :: "v"(lds_off), "v"((unsigned long long)gptr) : "memory");
}
__device__ __forceinline__ void wait_asynccnt0() {
  asm volatile("s_wait_asynccnt 0" ::: "memory");
}
__device__ __forceinline__ unsigned lds_off_of(const float* p) {
  // Generic pointer to LDS: low 32 bits are the LDS byte offset (aperture in high bits).
  return (unsigned)(unsigned long long)(const void*)p;
}

// ---------------------------------------------------------------------------
// Tall-skinny GEMM:  C[M, D] (+)= alpha * op(A) @ X,   D = 32 or 64, f32.
//   op=N : A is [M, K] row-major (lda = K-stride); A tile staged via
//          global_load_async_to_lds_b128 into [128][36] layout.
//   op=T : A is [K, M] row-major (lda = M-stride); A tile staged k-pair
//          interleaved: elem(k,m) -> lA[(k>>1)*288 + 2m + (k&1)].
// X is one/two [K,32] blocks, staged k-pair interleaved with stride SX so each
// WMMA B fragment (K=2s, K=2s+1 at N=n) is a single contiguous ds_load_b64.
// C accumulated with global f32 atomics (split-K over gridDim.y, pre-zeroed).
// Block: 256 threads (8 waves); wave w owns rows [blk*128 + 16w, +16) x D cols.
// ---------------------------------------------------------------------------
template <bool TRANS, int DCOLS>
__global__ __launch_bounds__(256) void gemm_ts(const float* __restrict__ A, int lda,
                                               const float* __restrict__ Xa,
                                               const float* __restrict__ Xb,
                                               float* __restrict__ Ca,
                                               float* __restrict__ Cb,
                                               float alpha, int kChunk) {
  constexpr int NT = DCOLS / 16;         // n-tiles per wave
  constexpr int SX = DCOLS * 2 + 32;     // X pair-row stride (floats), 32 mod 64 banks
  __shared__ float lA[4608];             // N: [128][36]   T: 16 pairs x 288
  __shared__ float lX[16 * (64 * 2 + 32)];

  const int tid = threadIdx.x;
  const int wv  = tid >> 5;
  const int lh  = tid & 15;
  const int sel = (tid >> 4) & 1;
  const int m0  = blockIdx.x * 128;
  const int kB  = blockIdx.y * kChunk;

  v8f acc[NT] = {};

  for (int k0 = kB; k0 < kB + kChunk; k0 += 32) {
    __syncthreads();
    // ---- stage A tile ----
    if (!TRANS) {
      #pragma unroll
      for (int r = 0; r < 4; ++r) {
        const int i = tid + 256 * r;                 // 1024 x 16B chunks
        const int row = i >> 3, c4 = (i & 7) << 2;   // [128][32] -> lA[128][36]
        async_g2l_b128(lds_off_of(&lA[row * 36 + c4]),
                       A + (size_t)(m0 + row) * lda + k0 + c4);
      }
    } else {
      #pragma unroll
      for (int r = 0; r < 4; ++r) {
        const int i = tid + 256 * r;
        const int krow = i >> 5, c4 = (i & 31) << 2; // [32][128] pair-interleave
        const float4 v = *(const float4*)(A + (size_t)(k0 + krow) * lda + m0 + c4);
        float* d = &lA[(krow >> 1) * 288 + (c4 << 1) + (krow & 1)];
        d[0] = v.x; d[2] = v.y; d[4] = v.z; d[6] = v.w;
      }
    }
    // ---- stage X tile (pair-interleaved) ----
    if (DCOLS == 64) {
      #pragma unroll
      for (int r = 0; r < 2; ++r) {
        const int i = tid + 256 * r;
        const int krow = i >> 4, col = (i & 15) << 2;
        const float* src = (col < 32) ? (Xa + (size_t)(k0 + krow) * 32 + col)
                                      : (Xb + (size_t)(k0 + krow) * 32 + (col - 32));
        const float4 v = *(const float4*)src;
        float* d = &lX[(krow >> 1) * SX + (col << 1) + (krow & 1)];
        d[0] = v.x; d[2] = v.y; d[4] = v.z; d[6] = v.w;
      }
    } else {
      const int krow = tid >> 3, col = (tid & 7) << 2;
      const float4 v = *(const float4*)(Xa + (size_t)(k0 + krow) * 32 + col);
      float* d = &lX[(krow >> 1) * SX + (col << 1) + (krow & 1)];
      d[0] = v.x; d[2] = v.y; d[4] = v.z; d[6] = v.w;
    }
    if (!TRANS) wait_asynccnt0();  // async LDS writes visible before barrier
    __syncthreads();

    // ---- 8 k-steps of 4, NT n-tiles; every fragment = one ds_load_b64 ----
    #pragma unroll
    for (int kk = 0; kk < 8; ++kk) {
      v2f a;
      if (!TRANS) {
        a = *(const v2f*)&lA[(wv * 16 + lh) * 36 + kk * 4 + 2 * sel];
      } else {
        a = *(const v2f*)&lA[(kk * 2 + sel) * 288 + ((wv * 16 + lh) << 1)];
      }
      #pragma unroll
      for (int nt = 0; nt < NT; ++nt) {
        const v2f b = *(const v2f*)&lX[(kk * 2 + sel) * SX + ((nt * 16 + lh) << 1)];
        acc[nt] = wmma_f32_16x16x4(a, b, acc[nt]);
      }
    }
  }

  // ---- epilogue: C frag VGPR j = C[M = 8*sel + j, N = lh]; atomic accumulate ----
  #pragma unroll
  for (int nt = 0; nt < NT; ++nt) {
    float* dst = (nt < 2) ? Ca : Cb;
    const int col = ((nt < 2) ? nt : nt - 2) * 16 + lh;
    #pragma unroll
    for (int j = 0; j < 8; ++j) {
      const int row = m0 + wv * 16 + sel * 8 + j;
      unsafeAtomicAdd(&dst[(size_t)row * 32 + col], alpha * acc[nt][j]);
    }
  }
}

// ---------------- elementwise glue ----------------
__global__ __launch_bounds__(256) void ew_add2(const float* __restrict__ a,
                                               const float* __restrict__ b,
                                               float* __restrict__ o, int n) {
  const int i = blockIdx.x * 256 + threadIdx.x;
  if (i < n) o[i] = a[i] + b[i];
}

__global__ __launch_bounds__(256) void ew_haxpy(const float* __restrict__ s,
                                                const float* __restrict__ x,
                                                float* __restrict__ o, int n) {
  const int i = blockIdx.x * 256 + threadIdx.x;
  if (i < n) o[i] = 0.5f * s[i] + x[i];
}

// ---------------- segment mean (G=8, D=32) ----------------
__global__ __launch_bounds__(256) void seg_accum(const float* __restrict__ h0,
                                                 const int* __restrict__ batch,
                                                 float* __restrict__ gsum,
                                                 float* __restrict__ gcnt) {
  __shared__ float lsum[256];  // [G=8][32]
  __shared__ float lcnt[8];
  const int tid = threadIdx.x;
  lsum[tid] = 0.0f;
  if (tid < 8) lcnt[tid] = 0.0f;
  __syncthreads();
  const int col = tid & 31;
  const int rl  = tid >> 5;
  const int base = blockIdx.x * 256;
  for (int rr = rl; rr < 256; rr += 8) {
    const int row = base + rr;
    const int g = batch[row] & 7;
    atomicAdd(&lsum[g * 32 + col], h0[(size_t)row * 32 + col]);
    if (col == 0) atomicAdd(&lcnt[g], 1.0f);
  }
  __syncthreads();
  unsafeAtomicAdd(&gsum[tid], lsum[tid]);
  if (tid < 8) unsafeAtomicAdd(&gcnt[tid], lcnt[tid]);
}

__global__ __launch_bounds__(256) void seg_final(const float* __restrict__ gsum,
                                                 const float* __restrict__ gcnt,
                                                 float* __restrict__ graph) {
  const int t = threadIdx.x;  // 256 = G*D
  graph[t] = gsum[t] / fmaxf(gcnt[t >> 5], 1.0f);
}

// ---------------------------------------------------------------------------
extern "C" void kernel_launch(void* const* d_in, const int* in_sizes, int n_in,
                              void* d_out, int out_size, void* d_ws, size_t ws_size,
                              hipStream_t stream) {
  (void)in_sizes; (void)n_in; (void)out_size; (void)ws_size;
  constexpr int N0 = 8192, N1 = 8192, N2 = 1024, D = 32;

  const float* B0   = (const float*)d_in[0];
  const float* B1   = (const float*)d_in[1];
  const float* A0   = (const float*)d_in[2];
  const float* A1   = (const float*)d_in[3];
  const float* coA1 = (const float*)d_in[4];
  const float* x0   = (const float*)d_in[5];
  const float* x1   = (const float*)d_in[6];
  const float* x2   = (const float*)d_in[7];
  const int*   batch = (const int*)d_in[8];

  float* out   = (float*)d_out;
  float* h0    = out;                          // [N0, D]
  float* h1    = h0 + (size_t)N0 * D;          // [N1, D]
  float* h2    = h1 + (size_t)N1 * D;          // [N2, D]
  float* graph = h2 + (size_t)N2 * D;          // [8,  D]

  float* ws = (float*)d_ws;
  const size_t SZ = (size_t)8192 * 32;
  float* y2h  = ws + 0 * SZ;  // 0.5 * B1 @ x2          [N1,32]
  float* t    = ws + 1 * SZ;  // B0 @ y2h               [N0,32]
  float* u2   = ws + 2 * SZ;  // B0 @ x1                [N0,32]
  float* wbuf = ws + 3 * SZ;  // A0^T @ t               [N0,32]
  float* sbuf = ws + 4 * SZ;  // B0^T @ (x0 + w)        [N1,32]
  float* e1   = ws + 5 * SZ;  // x0 + t                 [N0,32]
  float* e2   = ws + 6 * SZ;  // x0 + w                 [N0,32]
  float* q    = ws + 7 * SZ;  // 0.5*s + x1             [N1,32]
  float* gsum = ws + 8 * SZ;  // [8][32]
  float* gcnt = gsum + 256;   // [8]

  // Zero all atomic-accumulated targets.
  hipMemsetAsync(ws, 0, 5 * SZ * sizeof(float), stream);               // y2h,t,u2,w,s
  hipMemsetAsync(gsum, 0, (256 + 8) * sizeof(float), stream);
  hipMemsetAsync(h1, 0, ((size_t)N1 * D + (size_t)N2 * D) * sizeof(float), stream);

  const dim3 blk(256);
  // pass1: y2h = 0.5 * B1 @ x2                       (reads 33 MB)
  gemm_ts<false, 32><<<dim3(N1 / 128, 2), blk, 0, stream>>>(B1, N2, x2, nullptr, y2h, nullptr, 0.5f, N2 / 2);
  // pass2: [t | u2] = B0 @ [y2h | x1]                (reads 268 MB, one pass)
  gemm_ts<false, 64><<<dim3(N0 / 128, 4), blk, 0, stream>>>(B0, N1, y2h, x1, t, u2, 1.0f, N1 / 4);
  // h0 = t + u2   (A0^T x0 added by pass3's atomics)
  ew_add2<<<dim3((N0 * D) / 256), blk, 0, stream>>>(t, u2, h0, N0 * D);
  // pass3: [h0 += A0^T x0 | w = A0^T t]              (reads A0 once)
  gemm_ts<true, 64><<<dim3(N0 / 128, 4), blk, 0, stream>>>(A0, N0, x0, t, h0, wbuf, 1.0f, N0 / 4);
  // e1 = x0 + t ; e2 = x0 + w
  ew_add2<<<dim3((N0 * D) / 256), blk, 0, stream>>>(x0, t, e1, N0 * D);
  ew_add2<<<dim3((N0 * D) / 256), blk, 0, stream>>>(x0, wbuf, e2, N0 * D);
  // pass4: [h1 += B0^T e1 | s = B0^T e2]             (reads B0 once more)
  gemm_ts<true, 64><<<dim3(N1 / 128, 4), blk, 0, stream>>>(B0, N1, e1, e2, h1, sbuf, 1.0f, N0 / 4);
  // pass5: h1 += A1^T x1 ; h1 += coA1^T x1
  gemm_ts<true, 32><<<dim3(N1 / 128, 4), blk, 0, stream>>>(A1, N1, x1, nullptr, h1, nullptr, 1.0f, N1 / 4);
  gemm_ts<true, 32><<<dim3(N1 / 128, 4), blk, 0, stream>>>(coA1, N1, x1, nullptr, h1, nullptr, 1.0f, N1 / 4);
  // q = 0.5*s + x1 ; pass6: h2 += B1^T q
  ew_haxpy<<<dim3((N1 * D) / 256), blk, 0, stream>>>(sbuf, x1, q, N1 * D);
  gemm_ts<true, 32><<<dim3(N2 / 128, 16), blk, 0, stream>>>(B1, N2, q, nullptr, h2, nullptr, 1.0f, N1 / 16);
  // graph = segment-mean(h0, batch)
  seg_accum<<<dim3(N0 / 256), blk, 0, stream>>>(h0, batch, gsum, gcnt);
  seg_final<<<dim3(1), blk, 0, stream>>>(gsum, gcnt, graph);
}